// Tag_Transformer_18485539242621
// MI455X (gfx1250) — compile-verified
//
#include <hip/hip_runtime.h>
#include <hip/hip_bf16.h>
#include <math.h>

// ---------------- model constants (from reference) ----------------
#define Lc   4
#define Bc   32
#define Hc   8
#define Dc   512
#define DHc  64
#define FFc  2048
#define Sc   784
#define CAPc 1024

typedef __attribute__((ext_vector_type(2))) float v2f;
typedef __attribute__((ext_vector_type(8))) float v8f;

// ---------------------------------------------------------------------------
// Streaming float4 copy (grid-stride), with prefetch hint (global_prefetch_b8)
// ---------------------------------------------------------------------------
__global__ void copy_f4(const float4* __restrict__ src, float4* __restrict__ dst,
                        long n4) {
  long i = (long)blockIdx.x * blockDim.x + threadIdx.x;
  long stride = (long)gridDim.x * blockDim.x;
  for (; i < n4; i += stride) {
    if (i + stride < n4) __builtin_prefetch(src + i + stride, 0, 1);
    dst[i] = src[i];
  }
}

// ---------------------------------------------------------------------------
// WMMA f32 GEMM:  C[m][n] = sum_k A[m][k] * W[n][k] + bias[n]   (i.e. A @ W^T)
//   A: M x K row-major, W: N x K row-major, C: M x N row-major
//   grid = (N/16, M/16), block = 32 (one wave32; EXEC all ones around WMMA)
//   Uses V_WMMA_F32_16X16X4_F32, K-step 4.
//   A frag (16x4 f32): lanes 0-15  -> (A[m0+l][k0+0], A[m0+l][k0+1])
//                      lanes 16-31 -> (A[m0+l][k0+2], A[m0+l][k0+3])
//   B frag (4x16 f32) is W^T, mirrored layout: b = (W[n0+l][k0+2h], W[n0+l][k0+2h+1])
//   D layout: VGPR r, lanes 0-15 -> row r, col l; lanes 16-31 -> row r+8, col l
// ---------------------------------------------------------------------------
__global__ void wmma_gemm_xwT(const float* __restrict__ A,
                              const float* __restrict__ W,
                              const float* __restrict__ bias,
                              float* __restrict__ C,
                              int N, int K, int relu) {
  const int lane = threadIdx.x;
  const int half = lane >> 4;     // 0 or 1
  const int l    = lane & 15;
  const int n0   = blockIdx.x * 16;
  const int m0   = blockIdx.y * 16;

  const float* ap = A + (size_t)(m0 + l) * K + 2 * half;
  const float* bp = W + (size_t)(n0 + l) * K + 2 * half;

  v8f acc = {};
  for (int k0 = 0; k0 < K; k0 += 4) {
    v2f a; a.x = ap[k0]; a.y = ap[k0 + 1];
    v2f b; b.x = bp[k0]; b.y = bp[k0 + 1];
    acc = __builtin_amdgcn_wmma_f32_16x16x4_f32(
        /*neg_a=*/false, a, /*neg_b=*/false, b,
        /*c_mod=*/(short)0, acc, /*reuse_a=*/false, /*reuse_b=*/false);
  }

  const int n = n0 + l;
  const float bn = bias[n];
#pragma unroll
  for (int r = 0; r < 8; ++r) {
    int m = m0 + r + 8 * half;
    float v = acc[r] + bn;
    if (relu) v = fmaxf(v, 0.0f);
    C[(size_t)m * N + n] = v;
  }
}

// ---------------------------------------------------------------------------
// Scatter new k,v (from qkv buffer cols [512,1024) and [1024,1536)) into
// K_new/V_new at row t.  grid = 64, block = 256 (B*H*Dh = 16384 threads)
// ---------------------------------------------------------------------------
__global__ void scatter_kv(const float* __restrict__ qkv,
                           float* __restrict__ Ki, float* __restrict__ Vi,
                           const int* __restrict__ t_ptr) {
  int idx = blockIdx.x * blockDim.x + threadIdx.x;   // b*512 + h*64 + d
  int b  = idx >> 9;
  int hd = idx & 511;
  int t  = *t_ptr;
  size_t off = (((size_t)(b * Hc + (hd >> 6)) * CAPc) + t) * DHc + (hd & 63);
  Ki[off] = qkv[(size_t)b * (3 * Dc) + Dc + hd];
  Vi[off] = qkv[(size_t)b * (3 * Dc) + 2 * Dc + hd];
}

// ---------------------------------------------------------------------------
// Attention scores + softmax for one (b,h).  grid = B*H, block = 256.
//   Q[b*qrow + h*64 + d]; Kmat laid out (B,H,cap,64) contiguous.
//   use_mask: attend only where c >= *t_ptr (reference mask pos_cap >= t).
//   Masked rows are never read (saves ~t/CAP of K traffic).
// ---------------------------------------------------------------------------
__global__ void attn_softmax(const float* __restrict__ Q, int qrow,
                             const float* __restrict__ Kmat, int cap,
                             float* __restrict__ probs,
                             const int* __restrict__ t_ptr, int use_mask,
                             float scale) {
  __shared__ float sq[DHc];
  __shared__ float red[256];
  const int bh  = blockIdx.x;
  const int b   = bh >> 3;
  const int h   = bh & 7;
  const int tid = threadIdx.x;

  if (tid < DHc) sq[tid] = Q[(size_t)b * qrow + h * DHc + tid];
  __syncthreads();

  const int tcur = use_mask ? *t_ptr : 0;
  const float* Kb = Kmat + (size_t)bh * cap * DHc;

  float sc[4];
  float lmax = -INFINITY;
  const int nchunk = (cap + 255) / 256;   // <= 4
#pragma unroll
  for (int j = 0; j < 4; ++j) {
    float s = -INFINITY;
    int c = tid + j * 256;
    if (j < nchunk && c < cap && c >= tcur) {
      const float* kr = Kb + (size_t)c * DHc;
      float acc = 0.0f;
#pragma unroll
      for (int d = 0; d < DHc; d += 4) {
        float4 kv = *(const float4*)(kr + d);
        acc += sq[d] * kv.x + sq[d + 1] * kv.y + sq[d + 2] * kv.z + sq[d + 3] * kv.w;
      }
      s = acc * scale;
    }
    sc[j] = s;
    lmax = fmaxf(lmax, s);
  }

  red[tid] = lmax;
  __syncthreads();
  for (int off = 128; off > 0; off >>= 1) {
    if (tid < off) red[tid] = fmaxf(red[tid], red[tid + off]);
    __syncthreads();
  }
  const float gmax = red[0];
  __syncthreads();

  float lsum = 0.0f;
#pragma unroll
  for (int j = 0; j < 4; ++j) {
    float e = (sc[j] == -INFINITY) ? 0.0f : __expf(sc[j] - gmax);
    sc[j] = e;
    lsum += e;
  }
  red[tid] = lsum;
  __syncthreads();
  for (int off = 128; off > 0; off >>= 1) {
    if (tid < off) red[tid] += red[tid + off];
    __syncthreads();
  }
  const float inv = 1.0f / red[0];

#pragma unroll
  for (int j = 0; j < 4; ++j) {
    int c = tid + j * 256;
    if (j < nchunk && c < cap) probs[(size_t)bh * cap + c] = sc[j] * inv;
  }
}

// ---------------------------------------------------------------------------
// ctx[b][h*64+d] = sum_c probs[bh][c] * V[bh][c][d].  grid = B*H, block = 256
// (4 c-groups x 64 dims, LDS tree-reduce).  Skips masked rows entirely.
// ---------------------------------------------------------------------------
__global__ void attn_ctx(const float* __restrict__ probs,
                         const float* __restrict__ Vmat, int cap,
                         float* __restrict__ ctx,
                         const int* __restrict__ t_ptr, int use_mask) {
  __shared__ float part[4][DHc];
  const int bh  = blockIdx.x;
  const int b   = bh >> 3;
  const int h   = bh & 7;
  const int tid = threadIdx.x;
  const int g   = tid >> 6;
  const int d   = tid & 63;
  const int tcur = use_mask ? *t_ptr : 0;

  const float* Vb = Vmat + (size_t)bh * cap * DHc;
  const float* pb = probs + (size_t)bh * cap;

  float acc = 0.0f;
  for (int c = g; c < cap; c += 4)
    if (c >= tcur) acc += pb[c] * Vb[(size_t)c * DHc + d];
  part[g][d] = acc;
  __syncthreads();
  if (g == 0)
    ctx[(size_t)b * Dc + h * DHc + d] =
        part[0][d] + part[1][d] + part[2][d] + part[3][d];
}

// ---------------------------------------------------------------------------
// xb[b] = LayerNorm(xb[b] + y[b]) * g + beta.  grid = B, block = 256 (2 elems/thread)
// ---------------------------------------------------------------------------
__global__ void add_ln(float* __restrict__ xb, const float* __restrict__ y,
                       const float* __restrict__ g, const float* __restrict__ beta) {
  __shared__ float red[256];
  const int b   = blockIdx.x;
  const int tid = threadIdx.x;
  float v0 = xb[(size_t)b * Dc + tid]       + y[(size_t)b * Dc + tid];
  float v1 = xb[(size_t)b * Dc + 256 + tid] + y[(size_t)b * Dc + 256 + tid];

  red[tid] = v0 + v1;
  __syncthreads();
  for (int off = 128; off > 0; off >>= 1) {
    if (tid < off) red[tid] += red[tid + off];
    __syncthreads();
  }
  const float mean = red[0] * (1.0f / Dc);
  __syncthreads();

  const float d0 = v0 - mean, d1 = v1 - mean;
  red[tid] = d0 * d0 + d1 * d1;
  __syncthreads();
  for (int off = 128; off > 0; off >>= 1) {
    if (tid < off) red[tid] += red[tid + off];
    __syncthreads();
  }
  const float rstd = rsqrtf(red[0] * (1.0f / Dc) + 1e-5f);

  xb[(size_t)b * Dc + tid]       = d0 * rstd * g[tid]       + beta[tid];
  xb[(size_t)b * Dc + 256 + tid] = d1 * rstd * g[256 + tid] + beta[256 + tid];
}

// ---------------------------------------------------------------------------
extern "C" void kernel_launch(void* const* d_in, const int* in_sizes, int n_in,
                              void* d_out, int out_size, void* d_ws, size_t ws_size,
                              hipStream_t stream) {
  (void)in_sizes; (void)n_in; (void)out_size; (void)ws_size;

  const float* x       = (const float*)d_in[0];
  const float* sa_in_w = (const float*)d_in[1];
  const float* sa_in_b = (const float*)d_in[2];
  const float* sa_o_w  = (const float*)d_in[3];
  const float* sa_o_b  = (const float*)d_in[4];
  const float* q_w     = (const float*)d_in[5];
  const float* q_b     = (const float*)d_in[6];
  const float* ca_o_w  = (const float*)d_in[7];
  const float* ca_o_b  = (const float*)d_in[8];
  const float* K_mem   = (const float*)d_in[9];
  const float* V_mem   = (const float*)d_in[10];
  const float* K_buf   = (const float*)d_in[11];
  const float* V_buf   = (const float*)d_in[12];
  const float* w1      = (const float*)d_in[13];
  const float* b1      = (const float*)d_in[14];
  const float* w2      = (const float*)d_in[15];
  const float* b2      = (const float*)d_in[16];
  const float* ln1_w   = (const float*)d_in[17];
  const float* ln1_b   = (const float*)d_in[18];
  const float* ln2_w   = (const float*)d_in[19];
  const float* ln2_b   = (const float*)d_in[20];
  const float* ln3_w   = (const float*)d_in[21];
  const float* ln3_b   = (const float*)d_in[22];
  const int*   t_ptr   = (const int*)d_in[23];

  const size_t cacheN = (size_t)Lc * Bc * Hc * CAPc * DHc;   // 67,108,864
  const size_t memN_l = (size_t)Bc * Hc * Sc * DHc;          // per-layer mem KV
  const size_t cach_l = (size_t)Bc * Hc * CAPc * DHc;        // per-layer cache KV

  float* out_x = (float*)d_out;
  float* Kout  = out_x + (size_t)Bc * Dc;
  float* Vout  = Kout + cacheN;

  // workspace carving (floats)
  float* ws    = (float*)d_ws;
  float* xb    = ws;                 // B*D      = 16384
  float* qkv   = xb  + 16384;        // B*3D     = 49152
  float* ctx   = qkv + 49152;        // B*D      = 16384
  float* probs = ctx + 16384;        // B*H*CAP  = 262144
  float* tmp   = probs + 262144;     // B*D      = 16384
  float* q2    = tmp + 16384;        // B*D      = 16384
  float* ff1   = q2  + 16384;        // B*FF     = 65536

  const float scale = 0.125f;        // 1/sqrt(64)
  const dim3 blk32(32), blk256(256);

  // 1) bulk cache copy K_buf/V_buf -> K_new/V_new (dominant bandwidth term)
  copy_f4<<<4096, blk256, 0, stream>>>((const float4*)K_buf, (float4*)Kout,
                                       (long)(cacheN / 4));
  copy_f4<<<4096, blk256, 0, stream>>>((const float4*)V_buf, (float4*)Vout,
                                       (long)(cacheN / 4));
  // 2) xb = x
  copy_f4<<<16, blk256, 0, stream>>>((const float4*)x, (float4*)xb,
                                     (long)(Bc * Dc / 4));

  for (int i = 0; i < Lc; ++i) {
    const float* saw = sa_in_w + (size_t)i * 3 * Dc * Dc;
    const float* sab = sa_in_b + (size_t)i * 3 * Dc;
    const float* sow = sa_o_w  + (size_t)i * Dc * Dc;
    const float* sob = sa_o_b  + (size_t)i * Dc;
    const float* qw  = q_w     + (size_t)i * Dc * Dc;
    const float* qb  = q_b     + (size_t)i * Dc;
    const float* cow = ca_o_w  + (size_t)i * Dc * Dc;
    const float* cob = ca_o_b  + (size_t)i * Dc;
    const float* w1i = w1      + (size_t)i * FFc * Dc;
    const float* b1i = b1      + (size_t)i * FFc;
    const float* w2i = w2      + (size_t)i * Dc * FFc;
    const float* b2i = b2      + (size_t)i * Dc;
    const float* Kmi = K_mem   + (size_t)i * memN_l;
    const float* Vmi = V_mem   + (size_t)i * memN_l;
    float* Ki = Kout + (size_t)i * cach_l;
    float* Vi = Vout + (size_t)i * cach_l;

    // --- self-attention ---
    wmma_gemm_xwT<<<dim3(3 * Dc / 16, Bc / 16), blk32, 0, stream>>>(
        xb, saw, sab, qkv, 3 * Dc, Dc, 0);
    scatter_kv<<<64, blk256, 0, stream>>>(qkv, Ki, Vi, t_ptr);
    attn_softmax<<<Bc * Hc, blk256, 0, stream>>>(qkv, 3 * Dc, Ki, CAPc, probs,
                                                 t_ptr, 1, scale);
    attn_ctx<<<Bc * Hc, blk256, 0, stream>>>(probs, Vi, CAPc, ctx, t_ptr, 1);
    wmma_gemm_xwT<<<dim3(Dc / 16, Bc / 16), blk32, 0, stream>>>(
        ctx, sow, sob, tmp, Dc, Dc, 0);
    add_ln<<<Bc, blk256, 0, stream>>>(xb, tmp, ln1_w + (size_t)i * Dc,
                                      ln1_b + (size_t)i * Dc);

    // --- cross-attention ---
    wmma_gemm_xwT<<<dim3(Dc / 16, Bc / 16), blk32, 0, stream>>>(
        xb, qw, qb, q2, Dc, Dc, 0);
    attn_softmax<<<Bc * Hc, blk256, 0, stream>>>(q2, Dc, Kmi, Sc, probs,
                                                 t_ptr, 0, scale);
    attn_ctx<<<Bc * Hc, blk256, 0, stream>>>(probs, Vmi, Sc, ctx, t_ptr, 0);
    wmma_gemm_xwT<<<dim3(Dc / 16, Bc / 16), blk32, 0, stream>>>(
        ctx, cow, cob, tmp, Dc, Dc, 0);
    add_ln<<<Bc, blk256, 0, stream>>>(xb, tmp, ln2_w + (size_t)i * Dc,
                                      ln2_b + (size_t)i * Dc);

    // --- FFN ---
    wmma_gemm_xwT<<<dim3(FFc / 16, Bc / 16), blk32, 0, stream>>>(
        xb, w1i, b1i, ff1, FFc, Dc, 1);
    wmma_gemm_xwT<<<dim3(Dc / 16, Bc / 16), blk32, 0, stream>>>(
        ff1, w2i, b2i, tmp, Dc, FFc, 0);
    add_ln<<<Bc, blk256, 0, stream>>>(xb, tmp, ln3_w + (size_t)i * Dc,
                                      ln3_b + (size_t)i * Dc);
  }

  // final activations
  copy_f4<<<16, blk256, 0, stream>>>((const float4*)xb, (float4*)out_x,
                                     (long)(Bc * Dc / 4));
}